// PrototypeLayer_60756607369251
// MI455X (gfx1250) — compile-verified
//
#include <hip/hip_runtime.h>
#include <math.h>

typedef __attribute__((ext_vector_type(2))) float v2f;
typedef __attribute__((ext_vector_type(8))) float v8f;

#define NB      16
#define ND      128
#define NHW     3136            // 56*56
#define NP      2000
#define NTILE_P 125             // 2000/16
#define NCHUNK  14              // n-tiles per wave
#define NSPLIT  14              // 196/14 splits of the N dimension
#define WPB     8               // waves per block (256 threads, wave32)

// ---------------- init keys ----------------
__global__ void init_keys_kernel(unsigned long long* __restrict__ keys) {
    int i = blockIdx.x * blockDim.x + threadIdx.x;
    if (i < NB * NP) keys[i] = ~0ULL;
}

// ---------------- ||proto||^2 ----------------
__global__ void proto_sq_kernel(const float* __restrict__ proto, float* __restrict__ b2) {
    int p = blockIdx.x * blockDim.x + threadIdx.x;
    if (p >= NP) return;
    const float* r = proto + p * ND;
    float s = 0.f;
#pragma unroll 8
    for (int d = 0; d < ND; ++d) s += r[d] * r[d];
    b2[p] = s;
}

// ---------------- ||x||^2 per (b, n) ----------------
__global__ void x_sq_kernel(const float* __restrict__ x, float* __restrict__ a2) {
    int i = blockIdx.x * blockDim.x + threadIdx.x;
    if (i >= NB * NHW) return;
    int b = i / NHW, n = i % NHW;
    const float* xb = x + (size_t)b * ND * NHW + n;
    float s = 0.f;
#pragma unroll 8
    for (int d = 0; d < ND; ++d) { float v = xb[d * NHW]; s += v * v; }
    a2[i] = s;
}

// ------- fused WMMA fp32 GEMM + min/argmin key reduction -------
__global__ __launch_bounds__(WPB * 32)
void proto_gemm_min_kernel(const float* __restrict__ x,
                           const float* __restrict__ proto,
                           const float* __restrict__ a2,
                           const float* __restrict__ b2,
                           unsigned long long* __restrict__ keys) {
    const int lane = threadIdx.x & 31;
    const int wave = blockIdx.x * WPB + (threadIdx.x >> 5);

    const int wp = wave % NTILE_P;                 // which 16-prototype tile
    const int ws = (wave / NTILE_P) % NSPLIT;      // which N chunk
    const int wb = wave / (NTILE_P * NSPLIT);      // which batch

    const int col = lane & 15;                     // tile column (prototype)
    const int hi  = lane >> 4;                     // half-wave select
    const int p0  = wp * 16;

    const float* __restrict__ xb = x + (size_t)wb * ND * NHW;

    // Hoist the full B tile (16 protos x 128 K) into registers: contiguous b64 loads.
    // Lane layout for f32 WMMA B (4x16): lane<16 -> K={k0,k0+1}, lane>=16 -> K={k0+2,k0+3}.
    const float* __restrict__ bp = proto + (size_t)(p0 + col) * ND + 2 * hi;
    v2f breg[ND / 4];
#pragma unroll
    for (int k = 0; k < ND / 4; ++k)
        breg[k] = *(const v2f*)(bp + 4 * k);

    const float b2c = b2[p0 + col];

    unsigned long long mk[8];
#pragma unroll
    for (int v = 0; v < 8; ++v) mk[v] = ~0ULL;

    const int nbase = ws * NCHUNK * 16;
    for (int nt = 0; nt < NCHUNK; ++nt) {
        const int n0 = nbase + nt * 16;
        // A layout for f32 WMMA (16x4): lane<16 -> M=lane, K={k0,k0+1};
        // lane>=16 -> M=lane-16, K={k0+2,k0+3}. x is [D, N] per batch (stride NHW).
        const float* __restrict__ ap = xb + n0 + col + 2 * hi * NHW;

        v8f acc = {};
#pragma unroll
        for (int k = 0; k < ND / 4; ++k) {
            v2f a;
            a.x = ap[(4 * k)     * NHW];
            a.y = ap[(4 * k + 1) * NHW];
            acc = __builtin_amdgcn_wmma_f32_16x16x4_f32(
                /*neg_a=*/false, a, /*neg_b=*/false, breg[k],
                /*c_mod=*/(short)0, acc, /*reuse_a=*/false, /*reuse_b=*/false);
        }

        // Epilogue: d2 = ||a||^2 + ||b||^2 - 2ab, clamp >= 0; pack (d2_bits, n).
        const float* __restrict__ a2row = a2 + wb * NHW + n0;
#pragma unroll
        for (int v = 0; v < 8; ++v) {
            int m = v + hi * 8;                    // row within tile for this lane half
            float d2 = a2row[m] + b2c - 2.0f * acc[v];
            d2 = fmaxf(d2, 0.0f);
            unsigned long long key =
                ((unsigned long long)__float_as_uint(d2) << 32) | (unsigned)(n0 + m);
            mk[v] = key < mk[v] ? key : mk[v];
        }
    }

    unsigned long long best = mk[0];
#pragma unroll
    for (int v = 1; v < 8; ++v) best = mk[v] < best ? mk[v] : best;

    // Combine lane l with lane l^16 (same prototype column, other 8 rows).
    unsigned lo = (unsigned)best, hh = (unsigned)(best >> 32);
    unsigned lo2 = __shfl_xor(lo, 16, 32);
    unsigned hh2 = __shfl_xor(hh, 16, 32);
    unsigned long long other = ((unsigned long long)hh2 << 32) | lo2;
    if (other < best) best = other;

    if (lane < 16)
        atomicMin(&keys[(size_t)wb * NP + p0 + col], best);
}

// ---------------- finalize: decode keys, gather features ----------------
__global__ void finalize_kernel(const float* __restrict__ x,
                                const unsigned long long* __restrict__ keys,
                                float* __restrict__ out) {
    int bp = blockIdx.x;                 // (b, p) flat
    int b  = bp / NP;
    unsigned long long k = keys[bp];
    unsigned argmin_n = (unsigned)k;
    float d2 = __uint_as_float((unsigned)(k >> 32));
    float d  = sqrtf(d2);
    if (threadIdx.x == 0) {
        // similarity is monotone-decreasing in d -> max over n at the argmin distance
        out[bp]           = logf((d + 1.0f) / (d + 1e-7f));   // max_similarity [B,P]
        out[NB * NP + bp] = d;                                // min_distances  [B,200,10] (same flat order)
    }
    int t = threadIdx.x;                 // 0..127 feature channels
    out[2 * NB * NP + (size_t)bp * ND + t] =
        x[(size_t)b * ND * NHW + (size_t)t * NHW + argmin_n]; // min_features [B,P,D]
}

extern "C" void kernel_launch(void* const* d_in, const int* in_sizes, int n_in,
                              void* d_out, int out_size, void* d_ws, size_t ws_size,
                              hipStream_t stream) {
    (void)in_sizes; (void)n_in; (void)out_size; (void)ws_size;
    const float* x     = (const float*)d_in[0];   // [16,128,56,56]
    const float* proto = (const float*)d_in[1];   // [1,2000,128]
    float* out = (float*)d_out;

    // Workspace layout (~470 KB): keys | a2 | b2
    char* ws = (char*)d_ws;
    unsigned long long* keys = (unsigned long long*)ws;        // 16*2000*8 = 256000 B
    float* a2 = (float*)(ws + 262144);                         // 16*3136*4 = 200704 B
    float* b2 = (float*)(ws + 262144 + 204800);                // 2000*4    = 8000 B

    init_keys_kernel<<<(NB * NP + 255) / 256, 256, 0, stream>>>(keys);
    proto_sq_kernel<<<(NP + 127) / 128, 128, 0, stream>>>(proto, b2);
    x_sq_kernel<<<(NB * NHW + 255) / 256, 256, 0, stream>>>(x, a2);

    const int total_waves = NB * NSPLIT * NTILE_P;             // 28000
    proto_gemm_min_kernel<<<total_waves / WPB, WPB * 32, 0, stream>>>(x, proto, a2, b2, keys);

    finalize_kernel<<<NB * NP, ND, 0, stream>>>(x, keys, out);
}